// GumbelSigmoid_k_70428873720286
// MI455X (gfx1250) — compile-verified
//
#include <hip/hip_runtime.h>
#include <hip/hip_bf16.h>

#define NN   8192
#define DD   256
#define KK   32
#define EPSF 1e-6f
#define NEG_BIG (-3.402823466e38f)

typedef float v2f __attribute__((ext_vector_type(2)));
typedef float v8f __attribute__((ext_vector_type(8)));

// ---- flat output offsets (floats), reference return order ----
// edge_index [2, K*N] | edge_weights [K*N] | probs [N,N] | mask [N,N] | dist [N,N]
#define OFF_EDGE  ((size_t)0)
#define OFF_W     ((size_t)2 * KK * NN)            // 524288
#define OFF_PROBS (OFF_W + (size_t)KK * NN)        // 786432
#define OFF_MASK  (OFF_PROBS + (size_t)NN * NN)
#define OFF_DIST  (OFF_MASK + (size_t)NN * NN)

// ---------------------------------------------------------------------------
// Kernel 1: s[i] = sum_f x[i,f].  One wave (32 lanes) per row of 256 floats.
// ---------------------------------------------------------------------------
__global__ void rowsum_kernel(const float* __restrict__ x, float* __restrict__ s)
{
    const int wave = (blockIdx.x * blockDim.x + threadIdx.x) >> 5;  // row id
    const int lane = threadIdx.x & 31;
    const float* row = x + (size_t)wave * DD;
    float acc = 0.0f;
#pragma unroll
    for (int c = 0; c < DD; c += 32) acc += row[c + lane];
#pragma unroll
    for (int off = 16; off > 0; off >>= 1) acc += __shfl_xor(acc, off, 32);
    if (lane == 0) s[wave] = acc;
}

// ---------------------------------------------------------------------------
// Kernel 2: one V_WMMA_F32_16X16X4_F32 per 16x16 tile computes
//   dist[m,n] = s_r[m]^2 * 1  +  1 * s_c[n]^2  +  (-2 s_r[m]) * s_c[n]
// (rank-3 decomposition packed into K=0..2; K=3 is zero).
// A 16x4 f32 layout: VGPR0 = K0 (lanes 0-15) | K2 (lanes 16-31); VGPR1 = K1 | K3.
// B 4x16 f32 layout: VGPR0 = K0 row | K2 row;  VGPR1 = K1 row | K3 row.
// C/D: lanes 0-15 -> (M=r, N=lane); lanes 16-31 -> (M=8+r, N=lane-16).
// Then probs = exp(-t*dist)+eps in VALU; mask tile zeroed here too.
// ---------------------------------------------------------------------------
__global__ void fill_tiles_kernel(const float* __restrict__ s,
                                  const float* __restrict__ temp,
                                  float* __restrict__ probs,
                                  float* __restrict__ mask,
                                  float* __restrict__ dist)
{
    const int waveInBlk = threadIdx.x >> 5;
    const int lane      = threadIdx.x & 31;
    const int tile      = blockIdx.x * (blockDim.x >> 5) + waveInBlk; // 0..262143
    const int ti        = tile >> 9;        // tile row   (512 tiles)
    const int tj        = tile & 511;       // tile col
    const int m16       = lane & 15;
    const bool hi       = lane >= 16;

    const float srow = s[ti * 16 + m16];
    const float scol = s[tj * 16 + m16];

    v2f a, b;
    a.x = hi ? (-2.0f * srow) : (srow * srow);  // K2 | K0
    a.y = hi ? 0.0f           : 1.0f;           // K3 | K1
    b.x = hi ? scol           : 1.0f;           // K2 | K0
    b.y = hi ? 0.0f           : (scol * scol);  // K3 | K1

    v8f c = {};
    c = __builtin_amdgcn_wmma_f32_16x16x4_f32(
            /*neg_a=*/false, a, /*neg_b=*/false, b,
            /*c_mod=*/(short)0, c, /*reuse_a=*/false, /*reuse_b=*/false);

    const float t    = temp[0];
    const int   row0 = ti * 16 + (hi ? 8 : 0);
    const int   col  = tj * 16 + m16;
#pragma unroll
    for (int r = 0; r < 8; ++r) {
        const float  d    = c[r];
        const size_t addr = (size_t)(row0 + r) * NN + col;
        dist[addr]  = d;
        probs[addr] = __expf(-t * d) + EPSF;
        mask[addr]  = 0.0f;   // zero; top-k kernel scatters 1.0 afterwards
    }
}

// ---------------------------------------------------------------------------
// Kernel 3: per-column top-K of g[i,j] = log(exp(-t*d)+2eps) - log(-log(q+eps)).
// One lane per column j -> q reads coalesced across the wave for each row i.
// Top-K state in LDS [slot][tid] (bank-conflict free), threshold-gated
// replace-min. Ascending sort at the end to match argsort()[-K:] ordering.
// ---------------------------------------------------------------------------
#define TPB_TK 128
__global__ void topk_kernel(const float* __restrict__ q,
                            const float* __restrict__ s,
                            const float* __restrict__ temp,
                            float* __restrict__ edge,   // [2*K*N]
                            float* __restrict__ w,      // [K*N]
                            float* __restrict__ mask)   // [N*N], pre-zeroed
{
    __shared__ float tv[KK * TPB_TK];
    __shared__ int   tx[KK * TPB_TK];

    const int   tid = threadIdx.x;
    const int   j   = blockIdx.x * TPB_TK + tid;
    const float t   = temp[0];
    const float sj  = s[j];

#pragma unroll
    for (int k = 0; k < KK; ++k) {
        tv[k * TPB_TK + tid] = NEG_BIG;
        tx[k * TPB_TK + tid] = 0;
    }
    float curmin = NEG_BIG;

    for (int i = 0; i < NN; ++i) {
        // stream-prefetch q a few rows ahead (global_prefetch_b8)
        const int ipf = (i + 24 < NN) ? (i + 24) : (NN - 1);
        __builtin_prefetch(q + (size_t)ipf * NN + j, 0, 1);

        const float dsv = sj - s[i];
        const float dd  = dsv * dsv;
        const float pe  = __expf(-t * dd) + 2.0f * EPSF;     // probs + eps
        const float gq  = -__logf(q[(size_t)i * NN + j] + EPSF);
        const float g   = __logf(pe) - __logf(gq);

        if (g > curmin) {
            int   mslot = 0;
            float mval  = tv[tid];
            for (int k = 1; k < KK; ++k) {
                const float v = tv[k * TPB_TK + tid];
                if (v < mval) { mval = v; mslot = k; }
            }
            tv[mslot * TPB_TK + tid] = g;
            tx[mslot * TPB_TK + tid] = i;
            mval = tv[tid];
            for (int k = 1; k < KK; ++k) {
                const float v = tv[k * TPB_TK + tid];
                if (v < mval) mval = v;
            }
            curmin = mval;
        }
    }

    // ascending selection sort of the kept 32 (matches argsort order)
    for (int a = 0; a < KK - 1; ++a) {
        int   m  = a;
        float mv = tv[a * TPB_TK + tid];
        for (int b2 = a + 1; b2 < KK; ++b2) {
            const float v = tv[b2 * TPB_TK + tid];
            if (v < mv) { mv = v; m = b2; }
        }
        if (m != a) {
            const float fv = tv[a * TPB_TK + tid];
            tv[a * TPB_TK + tid] = tv[m * TPB_TK + tid];
            tv[m * TPB_TK + tid] = fv;
            const int iv = tx[a * TPB_TK + tid];
            tx[a * TPB_TK + tid] = tx[m * TPB_TK + tid];
            tx[m * TPB_TK + tid] = iv;
        }
    }

    for (int k = 0; k < KK; ++k) {
        const int    child = tx[k * TPB_TK + tid];
        const size_t e     = (size_t)k * NN + j;
        edge[e]                     = (float)child;   // child_nodes
        edge[(size_t)KK * NN + e]   = (float)j;       // parent_nodes
        w[e]                        = 1.0f;           // edge_weights
        mask[(size_t)child * NN + j] = 1.0f;          // symmetric scatter
        mask[(size_t)j * NN + child] = 1.0f;
    }
}

// ---------------------------------------------------------------------------
extern "C" void kernel_launch(void* const* d_in, const int* in_sizes, int n_in,
                              void* d_out, int out_size, void* d_ws, size_t ws_size,
                              hipStream_t stream)
{
    const float* x    = (const float*)d_in[0];   // [N, D]
    const float* q    = (const float*)d_in[1];   // [N, N]
    const float* temp = (const float*)d_in[2];   // [1]
    float*       out  = (float*)d_out;
    float*       s    = (float*)d_ws;            // [N] row sums (32 KB)

    // 1) row sums: 8192 rows, 8 waves/block
    rowsum_kernel<<<NN / 8, 256, 0, stream>>>(x, s);

    // 2) dense maps via WMMA tiles: 512x512 tiles, 8 waves(tiles)/block
    const int nTiles = (NN / 16) * (NN / 16);    // 262144
    fill_tiles_kernel<<<nTiles / 8, 256, 0, stream>>>(
        s, temp, out + OFF_PROBS, out + OFF_MASK, out + OFF_DIST);

    // 3) per-column top-K + edges + mask scatter
    topk_kernel<<<NN / TPB_TK, TPB_TK, 0, stream>>>(
        q, s, temp, out + OFF_EDGE, out + OFF_W, out + OFF_MASK);
}